// TemporalGNNEnsemble_62405874811497
// MI455X (gfx1250) — compile-verified
//
#include <hip/hip_runtime.h>

// ---------------------------------------------------------------------------
// TemporalGNNEnsemble forward for MI455X (gfx1250, wave32, WMMA).
//   N=50000 nodes, E=1.6M edges, IN=128, HID=64, OUT=16, TD=16, HDS=4 heads.
// Dense matmuls -> V_WMMA_F32_16X16X4_F32 (fp32-exact).
// Edge passes   -> float4 gathers + global_atomic_add_f32 scatters.
// Graph norm (deg / edge coeff) computed once, reused by all 4 GCN props.
// TGAT: len-1 attention == identity, so z = xt @ (Wv*Wo*gat_W) folded on device.
// ---------------------------------------------------------------------------

typedef float v2f __attribute__((ext_vector_type(2)));
typedef float v8f __attribute__((ext_vector_type(8)));

static const int N_  = 50000;
static const int E_  = 1600000;
static const int IN_ = 128;
static const int HID_ = 64;
static const int OUT_ = 16;
static const int TD_ = 16;

#define DEVFN __device__ __forceinline__

DEVFN float lrelu02(float x) { return x > 0.f ? x : 0.2f * x; }

DEVFN void atomicMaxF(float* a, float v) {
  // sign-split trick: works for mixed signs, avoids float atomicMax overload.
  if (v >= 0.f) atomicMax(reinterpret_cast<int*>(a), __float_as_int(v));
  else          atomicMin(reinterpret_cast<unsigned int*>(a), __float_as_uint(v));
}

#if defined(__gfx1250__) && __has_builtin(__builtin_amdgcn_wmma_f32_16x16x4_f32)
#define TGE_HAVE_WMMA_F32 1
#endif

// ---------------------------------------------------------------------------
// Generic WMMA GEMM: out[n,cout] = concat(A1[n,cin1], A2[n,cin2]) @ W[cin,cout]
//                                  (+ bias[cout]) (+ Cadd[n,cout])
// Block: 128 threads = 4 wave32; 64 rows per block; NT = cout/16 in {1,4}.
// A tile staged in LDS (stride 145 -> conflict-free); W read via cache (tiny).
// f32 WMMA fragment layout (ISA 7.12.2):
//   A 16x4 : lane l -> row l%16, vgpr v holds K = 2*(l/16) + v
//   B 4x16 : lane l -> col l%16, vgpr v holds K = 2*(l/16) + v
//   C/D    : vgpr r -> row r + 8*(l/16), col l%16
// ---------------------------------------------------------------------------
template <int NT>
__global__ __launch_bounds__(128) void tge_gemm(
    const float* __restrict__ A1, int cin1,
    const float* __restrict__ A2, int cin2,
    const float* __restrict__ W,
    const float* __restrict__ bias,
    const float* __restrict__ Cadd,
    float* __restrict__ out, int n) {
  const int cin  = cin1 + cin2;
  const int cout = NT * 16;
  __shared__ float As[64 * 145];

  const int row0 = blockIdx.x * 64;
  const int tid  = threadIdx.x;

  for (int i = tid; i < 64 * cin; i += 128) {
    int r = i / cin, c = i - r * cin;
    int gr = row0 + r;
    if (gr >= n) gr = n - 1;
    float v = (c < cin1) ? A1[(size_t)gr * cin1 + c]
                         : A2[(size_t)gr * cin2 + (c - cin1)];
    As[r * 145 + c] = v;
  }
  __syncthreads();

  const int wave  = tid >> 5;   // wave32
  const int lane  = tid & 31;
  const int lrow  = lane & 15;
  const int khalf = lane >> 4;
  const int koff  = khalf * 2;
  const int arow  = wave * 16 + lrow;

#ifdef TGE_HAVE_WMMA_F32
  v8f acc[NT];
#pragma unroll
  for (int t = 0; t < NT; ++t) acc[t] = v8f{};

  for (int kb = 0; kb < cin; kb += 4) {
    v2f a;
    a.x = As[arow * 145 + kb + koff];
    a.y = As[arow * 145 + kb + koff + 1];
    const float* w0 = W + (size_t)(kb + koff) * cout + lrow;
#pragma unroll
    for (int t = 0; t < NT; ++t) {
      v2f b;
      b.x = w0[t * 16];
      b.y = w0[cout + t * 16];
      acc[t] = __builtin_amdgcn_wmma_f32_16x16x4_f32(
          false, a, false, b, (short)0, acc[t], false, false);
    }
  }

#pragma unroll
  for (int t = 0; t < NT; ++t) {
    int col = t * 16 + lrow;
    float badd = bias ? bias[col] : 0.f;
#pragma unroll
    for (int r = 0; r < 8; ++r) {
      int grow = row0 + wave * 16 + khalf * 8 + r;
      if (grow < n) {
        float v = acc[t][r] + badd;
        if (Cadd) v += Cadd[(size_t)grow * cout + col];
        out[(size_t)grow * cout + col] = v;
      }
    }
  }
#else
  // Scalar fallback (keeps TU compiling if fp32 WMMA builtin is unavailable).
  (void)arow; (void)koff; (void)khalf; (void)lrow; (void)lane; (void)wave;
  for (int i = tid; i < 64 * cout; i += 128) {
    int r = i / cout, c = i - r * cout;
    int grow = row0 + r;
    if (grow >= n) continue;
    float s = bias ? bias[c] : 0.f;
    for (int k = 0; k < cin; ++k) s += As[r * 145 + k] * W[(size_t)k * cout + c];
    if (Cadd) s += Cadd[(size_t)grow * cout + c];
    out[(size_t)grow * cout + c] = s;
  }
#endif
}

// ------------------------------ small kernels ------------------------------

__global__ void tge_pnorm(const float* __restrict__ p, int cin, float* __restrict__ out) {
  __shared__ float red[256];
  float s = 0.f;
  for (int i = threadIdx.x; i < cin; i += 256) s += p[i] * p[i];
  red[threadIdx.x] = s;
  __syncthreads();
  for (int st = 128; st > 0; st >>= 1) {
    if (threadIdx.x < st) red[threadIdx.x] += red[threadIdx.x + st];
    __syncthreads();
  }
  if (threadIdx.x == 0) out[0] = sqrtf(red[0]);
}

__global__ void tge_scores(const float* __restrict__ h, const float* __restrict__ p,
                           const float* __restrict__ pnorm, int n, int cin,
                           float* __restrict__ sc) {
  int g = blockIdx.x * blockDim.x + threadIdx.x;
  if (g >= n) return;
  float s = 0.f;
  const float* hr = h + (size_t)g * cin;
  for (int k = 0; k < cin; ++k) s += hr[k] * p[k];
  sc[g] = s / (pnorm[0] + 1e-12f);
}

// Iterative top-64 argmax (descending, ties -> lower index), LDS bitmask.
__global__ void tge_topk64(const float* __restrict__ sc, int n, int* __restrict__ idx) {
  __shared__ unsigned mask[1568];     // supports n <= 50176
  __shared__ float bval[256];
  __shared__ int   bidx[256];
  int tid = threadIdx.x;
  int words = (n + 31) >> 5;
  for (int i = tid; i < words; i += 256) mask[i] = 0u;
  __syncthreads();
  for (int it = 0; it < 64; ++it) {
    float best = -3.4e38f;
    int bi = 0x7fffffff;
    for (int i = tid; i < n; i += 256) {
      if ((mask[i >> 5] >> (i & 31)) & 1u) continue;
      float v = sc[i];
      if (v > best || (v == best && i < bi)) { best = v; bi = i; }
    }
    bval[tid] = best; bidx[tid] = bi;
    __syncthreads();
    for (int st = 128; st > 0; st >>= 1) {
      if (tid < st) {
        float v = bval[tid + st]; int j = bidx[tid + st];
        if (v > bval[tid] || (v == bval[tid] && j < bidx[tid])) {
          bval[tid] = v; bidx[tid] = j;
        }
      }
      __syncthreads();
    }
    if (tid == 0) {
      idx[it] = bidx[0];
      mask[bidx[0] >> 5] |= 1u << (bidx[0] & 31);
    }
    __syncthreads();
  }
}

__global__ void tge_build_xt(const float* __restrict__ h, const float* __restrict__ sc,
                             const int* __restrict__ idx, int cin, float* __restrict__ Xt) {
  int g = blockIdx.x * blockDim.x + threadIdx.x;
  if (g >= cin * 64) return;
  int i = g >> 6, j = g & 63;
  int nd = idx[j];
  Xt[i * 64 + j] = tanhf(sc[nd]) * h[(size_t)nd * cin + i];
}

__global__ void tge_gru_zr(const float* __restrict__ Wz, const float* __restrict__ Uz,
                           const float* __restrict__ bz, const float* __restrict__ Wr,
                           const float* __restrict__ Ur, const float* __restrict__ br,
                           const float* __restrict__ Xt, const float* __restrict__ H0,
                           int cin, float* __restrict__ Zb, float* __restrict__ RHb) {
  int g = blockIdx.x * blockDim.x + threadIdx.x;
  if (g >= cin * 64) return;
  int i = g >> 6, j = g & 63;
  float az = bz[i], ar = br[i];
  for (int k = 0; k < cin; ++k) {
    float xt = Xt[k * 64 + j], h0 = H0[k * 64 + j];
    az += Wz[i * cin + k] * xt + Uz[i * cin + k] * h0;
    ar += Wr[i * cin + k] * xt + Ur[i * cin + k] * h0;
  }
  float Z = 1.f / (1.f + expf(-az));
  float R = 1.f / (1.f + expf(-ar));
  Zb[g]  = Z;
  RHb[g] = R * H0[g];
}

__global__ void tge_gru_h(const float* __restrict__ Wh, const float* __restrict__ Uh,
                          const float* __restrict__ bh, const float* __restrict__ Xt,
                          const float* __restrict__ RH, const float* __restrict__ H0,
                          const float* __restrict__ Zb, int cin, float* __restrict__ Wl) {
  int g = blockIdx.x * blockDim.x + threadIdx.x;
  if (g >= cin * 64) return;
  int i = g >> 6, j = g & 63;
  float ah = bh[i];
  for (int k = 0; k < cin; ++k)
    ah += Wh[i * cin + k] * Xt[k * 64 + j] + Uh[i * cin + k] * RH[k * 64 + j];
  float Ht = tanhf(ah);
  float Z = Zb[g];
  Wl[g] = (1.f - Z) * H0[g] + Z * Ht;
}

__global__ void tge_mm64(const float* __restrict__ A, const float* __restrict__ B,
                         float* __restrict__ C) {
  int g = blockIdx.x * blockDim.x + threadIdx.x;
  if (g >= 64 * 64) return;
  int i = g >> 6, j = g & 63;
  float s = 0.f;
  for (int k = 0; k < 64; ++k) s += A[i * 64 + k] * B[k * 64 + j];
  C[g] = s;
}

__global__ void tge_te(const float* __restrict__ ts, const float* __restrict__ wt,
                       const float* __restrict__ bt, int n, float* __restrict__ te) {
  int g = blockIdx.x * blockDim.x + threadIdx.x;
  if (g >= n * TD_) return;
  int nd = g >> 4, k = g & 15;
  te[g] = cosf(ts[nd] * wt[k] + bt[k]);
}

// ------------------------------ graph norm ---------------------------------

__global__ void tge_deg_init(float* __restrict__ deg, int n) {
  int g = blockIdx.x * blockDim.x + threadIdx.x;
  if (g < n) deg[g] = 1.f;   // self loop weight 1
}
__global__ void tge_deg_edge(const int* __restrict__ dst, const float* __restrict__ ew,
                             int e, float* __restrict__ deg) {
  int g = blockIdx.x * blockDim.x + threadIdx.x;
  if (g < e) atomicAdd(&deg[dst[g]], ew[g]);
}
__global__ void tge_cnorm(const int* __restrict__ src, const int* __restrict__ dst,
                          const float* __restrict__ ew, const float* __restrict__ deg,
                          int e, float* __restrict__ cn) {
  int g = blockIdx.x * blockDim.x + threadIdx.x;
  if (g < e) cn[g] = ew[g] * rsqrtf(deg[src[g]] * deg[dst[g]] + 1e-12f);
}
__global__ void tge_selfc(const float* __restrict__ deg, int n, float* __restrict__ sc) {
  int g = blockIdx.x * blockDim.x + threadIdx.x;
  if (g < n) { float d = deg[g]; sc[g] = rsqrtf(d * d + 1e-12f); }
}

__global__ void tge_prop_init(const float* __restrict__ hin, const float* __restrict__ selfc,
                              float* __restrict__ hout, int n64) {
  int g = blockIdx.x * blockDim.x + threadIdx.x;
  if (g >= n64) return;
  hout[g] = selfc[g >> 6] * hin[g];
}
__global__ void tge_prop_edge(const int* __restrict__ src, const int* __restrict__ dst,
                              const float* __restrict__ cn, const float* __restrict__ hin,
                              float* __restrict__ hout, int e) {
  int g = blockIdx.x * blockDim.x + threadIdx.x;
  if (g >= e * 16) return;
  int ed = g >> 4, q = g & 15, c = q * 4;
  int s = src[ed], d = dst[ed];
  float w = cn[ed];
  float4 v = *reinterpret_cast<const float4*>(hin + (size_t)s * 64 + c);
  float* o = hout + (size_t)d * 64 + c;
  atomicAdd(o + 0, w * v.x);
  atomicAdd(o + 1, w * v.y);
  atomicAdd(o + 2, w * v.z);
  atomicAdd(o + 3, w * v.w);
}

__global__ void tge_relu(float* __restrict__ h, int n) {
  int g = blockIdx.x * blockDim.x + threadIdx.x;
  if (g < n) h[g] = fmaxf(h[g], 0.f);
}

// ------------------------------ GAT ----------------------------------------

__global__ void tge_esed(const float* __restrict__ z, const float* __restrict__ a_s,
                         const float* __restrict__ a_d, int n,
                         float* __restrict__ es, float* __restrict__ ed) {
  int g = blockIdx.x * blockDim.x + threadIdx.x;
  if (g >= n * 4) return;
  int nd = g >> 2, hd = g & 3;
  const float* zr = z + (size_t)nd * 64 + hd * 16;
  const float* as = a_s + hd * 16;
  const float* ad = a_d + hd * 16;
  float s = 0.f, t = 0.f;
  for (int f = 0; f < 16; ++f) { s += zr[f] * as[f]; t += zr[f] * ad[f]; }
  es[g] = s; ed[g] = t;
}
__global__ void tge_gat_minit(const float* __restrict__ es, const float* __restrict__ ed,
                              int n4, float* __restrict__ m) {
  int g = blockIdx.x * blockDim.x + threadIdx.x;
  if (g < n4) m[g] = lrelu02(es[g] + ed[g]);   // self-loop logit
}
__global__ void tge_gat_max(const int* __restrict__ src, const int* __restrict__ dst,
                            const float* __restrict__ es, const float* __restrict__ ed,
                            int e, float* __restrict__ m) {
  int g = blockIdx.x * blockDim.x + threadIdx.x;
  if (g >= e) return;
  int s = src[g], d = dst[g];
  for (int hd = 0; hd < 4; ++hd)
    atomicMaxF(&m[d * 4 + hd], lrelu02(es[s * 4 + hd] + ed[d * 4 + hd]));
}
__global__ void tge_gat_dinit(const float* __restrict__ es, const float* __restrict__ ed,
                              const float* __restrict__ m, int n4, float* __restrict__ dn) {
  int g = blockIdx.x * blockDim.x + threadIdx.x;
  if (g < n4) dn[g] = expf(lrelu02(es[g] + ed[g]) - m[g]);   // self-loop term
}
__global__ void tge_gat_dsum(const int* __restrict__ src, const int* __restrict__ dst,
                             const float* __restrict__ es, const float* __restrict__ ed,
                             const float* __restrict__ m, int e, float* __restrict__ dn) {
  int g = blockIdx.x * blockDim.x + threadIdx.x;
  if (g >= e) return;
  int s = src[g], d = dst[g];
  for (int hd = 0; hd < 4; ++hd)
    atomicAdd(&dn[d * 4 + hd], expf(lrelu02(es[s * 4 + hd] + ed[d * 4 + hd]) - m[d * 4 + hd]));
}
__global__ void tge_gat_oinit(const float* __restrict__ z, const float* __restrict__ es,
                              const float* __restrict__ ed, const float* __restrict__ m,
                              const float* __restrict__ dn, int n, float* __restrict__ out) {
  int g = blockIdx.x * blockDim.x + threadIdx.x;
  if (g >= n * 64) return;
  int nd = g >> 6, c = g & 63, hd = c >> 4;
  int q = nd * 4 + hd;
  float a = expf(lrelu02(es[q] + ed[q]) - m[q]) / (dn[q] + 1e-12f);
  out[g] = a * z[g];
}
__global__ void tge_gat_edge(const int* __restrict__ src, const int* __restrict__ dst,
                             const float* __restrict__ z, const float* __restrict__ es,
                             const float* __restrict__ ed, const float* __restrict__ m,
                             const float* __restrict__ dn, int e, float* __restrict__ out) {
  int g = blockIdx.x * blockDim.x + threadIdx.x;
  if (g >= e * 16) return;
  int eid = g >> 4, q = g & 15, c = q * 4, hd = q >> 2;
  int s = src[eid], d = dst[eid];
  float a = expf(lrelu02(es[s * 4 + hd] + ed[d * 4 + hd]) - m[d * 4 + hd]) /
            (dn[d * 4 + hd] + 1e-12f);
  float4 v = *reinterpret_cast<const float4*>(z + (size_t)s * 64 + c);
  float* o = out + (size_t)d * 64 + c;
  atomicAdd(o + 0, a * v.x);
  atomicAdd(o + 1, a * v.y);
  atomicAdd(o + 2, a * v.z);
  atomicAdd(o + 3, a * v.w);
}

__global__ void tge_addbias(float* __restrict__ h, const float* __restrict__ b,
                            int total, int c) {
  int g = blockIdx.x * blockDim.x + threadIdx.x;
  if (g < total) h[g] += b[g % c];
}

__global__ void tge_ensemble(const float* __restrict__ o1, const float* __restrict__ o2,
                             const float* __restrict__ o3, const float* __restrict__ ensw,
                             float* __restrict__ out, int n) {
  int g = blockIdx.x * blockDim.x + threadIdx.x;
  if (g >= n) return;
  float a = ensw[0], b = ensw[1], c = ensw[2];
  float mx = fmaxf(a, fmaxf(b, c));
  float ea = expf(a - mx), eb = expf(b - mx), ec = expf(c - mx);
  float s = ea + eb + ec;
  out[g] = (ea * o1[g] + eb * o2[g] + ec * o3[g]) / s;
}

// ---------------------------------------------------------------------------
// Host orchestration. Param flattening assumed in setup_inputs() insertion
// order (recursively):
//  0 x, 1 edge_index, 2 timestamps, 3 edge_weight,
//  egcn L0: 4 p,5 W0,6 Wz,7 Uz,8 bz,9 Wr,10 Ur,11 br,12 Wh,13 Uh,14 bh
//  egcn L1: 15..25 (same order), 26 cls_W, 27 cls_b
//  tgat: 28 wt, 29 bt, 30 Wp, 31 bp,
//        L0: 32 Wt,33 Wv,34 Wo,35 gat_W,36 a_s,37 a_d,38 gat_b, L1: 39..45,
//        46 cls_W, 47 cls_b
//  tgn: 48 wt, 49 bt, 50 W0, 51 b0, 52 W1, 53 b1, 54 cls_W, 55 cls_b
//  56 ens_w
// ---------------------------------------------------------------------------
extern "C" void kernel_launch(void* const* d_in, const int* in_sizes, int n_in,
                              void* d_out, int out_size, void* d_ws, size_t ws_size,
                              hipStream_t stream) {
  (void)in_sizes; (void)out_size;
  if (n_in < 57) return;

  const float* x   = (const float*)d_in[0];
  const int*   ei  = (const int*)d_in[1];
  const float* ts  = (const float*)d_in[2];
  const float* ew  = (const float*)d_in[3];
  const int* src = ei;
  const int* dst = ei + E_;
  auto P = [&](int i) { return (const float*)d_in[i]; };

  // workspace layout (floats)
  float* w = (float*)d_ws;
  size_t off = 0;
  auto alloc = [&](size_t cnt) { float* p = w + off; off += cnt; return p; };
  float* TE1 = alloc((size_t)N_ * 16);
  float* TE2 = alloc((size_t)N_ * 16);
  float* CN  = alloc(E_);
  float* DEG = alloc(N_);
  float* SFC = alloc(N_);
  float* SCR = alloc(N_);
  float* O1  = alloc((size_t)N_ * 16);
  float* O2  = alloc((size_t)N_ * 16);
  float* O3  = alloc((size_t)N_ * 16);
  float* B0  = alloc((size_t)N_ * 64);
  float* B1  = alloc((size_t)N_ * 64);
  float* B2  = alloc((size_t)N_ * 64);
  float* ES  = alloc((size_t)N_ * 4);
  float* ED  = alloc((size_t)N_ * 4);
  float* MM  = alloc((size_t)N_ * 4);
  float* DN  = alloc((size_t)N_ * 4);
  float* XT  = alloc(128 * 64);
  float* ZB  = alloc(128 * 64);
  float* RH  = alloc(128 * 64);
  float* WL  = alloc(128 * 64);
  float* T64 = alloc(64 * 64);
  float* MZ  = alloc(64 * 64);
  float* PN  = alloc(1);
  int*   IDX = (int*)alloc(64);
  if (ws_size < off * sizeof(float)) return;

  auto nb = [](long total) { return (unsigned)((total + 255) / 256); };
  const int gblocks = (N_ + 63) / 64;

  auto gemm = [&](const float* A1, int c1, const float* A2, int c2,
                  const float* Wm, const float* bias, const float* Cadd,
                  float* outp, int cout) {
    if (cout == 64)
      tge_gemm<4><<<gblocks, 128, 0, stream>>>(A1, c1, A2, c2, Wm, bias, Cadd, outp, N_);
    else
      tge_gemm<1><<<gblocks, 128, 0, stream>>>(A1, c1, A2, c2, Wm, bias, Cadd, outp, N_);
  };
  auto prop = [&](const float* hin, float* hout) {
    tge_prop_init<<<nb((long)N_ * 64), 256, 0, stream>>>(hin, SFC, hout, N_ * 64);
    tge_prop_edge<<<nb((long)E_ * 16), 256, 0, stream>>>(src, dst, CN, hin, hout, E_);
  };

  // shared symmetric GCN normalization (same for all 4 props)
  tge_deg_init<<<nb(N_), 256, 0, stream>>>(DEG, N_);
  tge_deg_edge<<<nb(E_), 256, 0, stream>>>(dst, ew, E_, DEG);
  tge_cnorm<<<nb(E_), 256, 0, stream>>>(src, dst, ew, DEG, E_, CN);
  tge_selfc<<<nb(N_), 256, 0, stream>>>(DEG, N_, SFC);

  // -------------------- EvolveGCN-H --------------------
  {  // layer 0: cin=128, h=x
    tge_pnorm<<<1, 256, 0, stream>>>(P(4), 128, PN);
    tge_scores<<<nb(N_), 256, 0, stream>>>(x, P(4), PN, N_, 128, SCR);
    tge_topk64<<<1, 256, 0, stream>>>(SCR, N_, IDX);
    tge_build_xt<<<nb(128 * 64), 256, 0, stream>>>(x, SCR, IDX, 128, XT);
    tge_gru_zr<<<nb(128 * 64), 256, 0, stream>>>(P(6), P(7), P(8), P(9), P(10), P(11),
                                                 XT, P(5), 128, ZB, RH);
    tge_gru_h<<<nb(128 * 64), 256, 0, stream>>>(P(12), P(13), P(14), XT, RH, P(5), ZB,
                                                128, WL);
    gemm(x, 128, nullptr, 0, WL, nullptr, nullptr, B0, 64);
    prop(B0, B1);
    tge_relu<<<nb((long)N_ * 64), 256, 0, stream>>>(B1, N_ * 64);
  }
  {  // layer 1: cin=64, h=B1
    tge_pnorm<<<1, 256, 0, stream>>>(P(15), 64, PN);
    tge_scores<<<nb(N_), 256, 0, stream>>>(B1, P(15), PN, N_, 64, SCR);
    tge_topk64<<<1, 256, 0, stream>>>(SCR, N_, IDX);
    tge_build_xt<<<nb(64 * 64), 256, 0, stream>>>(B1, SCR, IDX, 64, XT);
    tge_gru_zr<<<nb(64 * 64), 256, 0, stream>>>(P(17), P(18), P(19), P(20), P(21), P(22),
                                                XT, P(16), 64, ZB, RH);
    tge_gru_h<<<nb(64 * 64), 256, 0, stream>>>(P(23), P(24), P(25), XT, RH, P(16), ZB,
                                               64, WL);
    gemm(B1, 64, nullptr, 0, WL, nullptr, nullptr, B0, 64);
    prop(B0, B2);
    tge_relu<<<nb((long)N_ * 64), 256, 0, stream>>>(B2, N_ * 64);
  }
  gemm(B2, 64, nullptr, 0, P(26), P(27), nullptr, O1, 16);

  // -------------------- TGAT --------------------
  tge_te<<<nb((long)N_ * 16), 256, 0, stream>>>(ts, P(28), P(29), N_, TE1);
  gemm(x, 128, TE1, 16, P(30), P(31), nullptr, B0, 64);  // h = [x|te]@Wp + bp
  for (int l = 0; l < 2; ++l) {
    int b = 32 + l * 7;  // Wt,Wv,Wo,gat_W,a_s,a_d,gat_b
    tge_mm64<<<nb(64 * 64), 256, 0, stream>>>(P(b + 1), P(b + 2), T64);  // Wv@Wo
    tge_mm64<<<nb(64 * 64), 256, 0, stream>>>(T64, P(b + 3), MZ);        // @gat_W
    gemm(TE1, 16, nullptr, 0, P(b + 0), nullptr, B0, B1, 64);            // xt = h + te@Wt
    gemm(B1, 64, nullptr, 0, MZ, nullptr, nullptr, B2, 64);              // z
    tge_esed<<<nb((long)N_ * 4), 256, 0, stream>>>(B2, P(b + 4), P(b + 5), N_, ES, ED);
    tge_gat_minit<<<nb((long)N_ * 4), 256, 0, stream>>>(ES, ED, N_ * 4, MM);
    tge_gat_max<<<nb(E_), 256, 0, stream>>>(src, dst, ES, ED, E_, MM);
    tge_gat_dinit<<<nb((long)N_ * 4), 256, 0, stream>>>(ES, ED, MM, N_ * 4, DN);
    tge_gat_dsum<<<nb(E_), 256, 0, stream>>>(src, dst, ES, ED, MM, E_, DN);
    tge_gat_oinit<<<nb((long)N_ * 64), 256, 0, stream>>>(B2, ES, ED, MM, DN, N_, B0);
    tge_gat_edge<<<nb((long)E_ * 16), 256, 0, stream>>>(src, dst, B2, ES, ED, MM, DN, E_, B0);
    tge_addbias<<<nb((long)N_ * 64), 256, 0, stream>>>(B0, P(b + 6), N_ * 64, 64);
  }
  gemm(B0, 64, nullptr, 0, P(46), P(47), nullptr, O2, 16);

  // -------------------- TGN --------------------
  tge_te<<<nb((long)N_ * 16), 256, 0, stream>>>(ts, P(48), P(49), N_, TE2);
  gemm(x, 128, TE2, 16, P(50), P(51), nullptr, B1, 64);
  prop(B1, B2);
  tge_relu<<<nb((long)N_ * 64), 256, 0, stream>>>(B2, N_ * 64);
  gemm(B2, 64, TE2, 16, P(52), P(53), nullptr, B1, 64);
  prop(B1, B0);
  tge_relu<<<nb((long)N_ * 64), 256, 0, stream>>>(B0, N_ * 64);
  gemm(B0, 64, nullptr, 0, P(54), P(55), nullptr, O3, 16);

  // -------------------- ensemble --------------------
  tge_ensemble<<<nb((long)N_ * 16), 256, 0, stream>>>(O1, O2, O3, P(56),
                                                      (float*)d_out, N_ * 16);
}